// RuleConvolutionLayer_44006234915593
// MI455X (gfx1250) — compile-verified
//
#include <hip/hip_runtime.h>
#include <stdint.h>

// Rule-indexed sparse graph convolution for MI455X (gfx1250, wave32).
// Pipeline: device CSR build (histogram -> scan -> scatter of (dst, w4)),
// then one wave32 per node:
//   - async global->LDS staging of edge records (GLOBAL_LOAD_ASYNC_TO_LDS_*,
//     ASYNCcnt double buffering)
//   - per-node KxF GEMM over edges via V_WMMA_F32_16X16X4_F32:
//       A(16x4)=rule weights (rows 0..3 = channels), B(4x16)=x[dst] features,
//       C/D rows 0..3 = out[k, feature-tile], bias-seeded.

#define KCH 4
#define LBL 50
#define PRP 16
#define FDIM 128
#define FV4 32                 // FDIM / 4
#define LLP (LBL * LBL * PRP)  // 40000

typedef __attribute__((ext_vector_type(2))) float v2f;
typedef __attribute__((ext_vector_type(8))) float v8f;

// Only meaningful to check on the device pass; the host pass has no AMDGPU builtins.
#if defined(__HIP_DEVICE_COMPILE__)
#if !__has_builtin(__builtin_amdgcn_wmma_f32_16x16x4_f32)
#error "device pass: __builtin_amdgcn_wmma_f32_16x16x4_f32 not available"
#endif
#endif

// ---------- CDNA5 async-LDS helpers (inline asm per CDNA5 ISA) ----------
__device__ __forceinline__ void async_load_lds_b128(uint32_t lds_byte_addr, const void* gaddr) {
  asm volatile("global_load_async_to_lds_b128 %0, %1, off"
               :: "v"(lds_byte_addr), "v"((unsigned long long)(uintptr_t)gaddr)
               : "memory");
}
__device__ __forceinline__ void async_load_lds_b32(uint32_t lds_byte_addr, const void* gaddr) {
  asm volatile("global_load_async_to_lds_b32 %0, %1, off"
               :: "v"(lds_byte_addr), "v"((unsigned long long)(uintptr_t)gaddr)
               : "memory");
}
__device__ __forceinline__ void wait_async_le2() {
#if __has_builtin(__builtin_amdgcn_s_wait_asynccnt)
  __builtin_amdgcn_s_wait_asynccnt(2);
#else
  asm volatile("s_wait_asynccnt 0x2" ::: "memory");
#endif
}
__device__ __forceinline__ void wait_async_le0() {
#if __has_builtin(__builtin_amdgcn_s_wait_asynccnt)
  __builtin_amdgcn_s_wait_asynccnt(0);
#else
  asm volatile("s_wait_asynccnt 0x0" ::: "memory");
#endif
}
__device__ __forceinline__ uint32_t lds_addr_of(const void* p) {
  // Generic pointer to __shared__: low 32 bits are the LDS byte address.
  return (uint32_t)(uintptr_t)p;
}

// ---------- CSR build ----------
__global__ void k_zero_u32(uint32_t* __restrict__ p, int n) {
  int i = blockIdx.x * blockDim.x + threadIdx.x;
  if (i < n) p[i] = 0u;
}

__global__ void k_hist(const int* __restrict__ src, uint32_t* __restrict__ cnt, int E) {
  int i = blockIdx.x * blockDim.x + threadIdx.x;
  if (i < E) atomicAdd(&cnt[src[i]], 1u);
}

__global__ __launch_bounds__(1024) void k_scan(const uint32_t* __restrict__ cnt,
                                               uint32_t* __restrict__ offs,
                                               uint32_t* __restrict__ cursor,
                                               int N, int E) {
  __shared__ uint32_t tmp[1024];
  __shared__ uint32_t carry;
  const int tid = threadIdx.x;
  if (tid == 0) carry = 0u;
  __syncthreads();
  for (int base = 0; base < N; base += 1024) {
    int idx = base + tid;
    uint32_t v = (idx < N) ? cnt[idx] : 0u;
    tmp[tid] = v;
    __syncthreads();
    for (int off = 1; off < 1024; off <<= 1) {
      uint32_t t = (tid >= off) ? tmp[tid - off] : 0u;
      __syncthreads();
      tmp[tid] += t;
      __syncthreads();
    }
    uint32_t excl = tmp[tid] - v;
    uint32_t c = carry;
    if (idx < N) { offs[idx] = c + excl; cursor[idx] = c + excl; }
    __syncthreads();
    if (tid == 0) carry = c + tmp[1023];
    __syncthreads();
  }
  if (tid == 0) offs[N] = (uint32_t)E;
}

__global__ void k_scatter(const int* __restrict__ src, const int* __restrict__ dst,
                          const int* __restrict__ prop, const int* __restrict__ lab,
                          const float* __restrict__ W,
                          uint32_t* __restrict__ cursor,
                          int* __restrict__ sdst, float4* __restrict__ sw4, int E) {
  int e = blockIdx.x * blockDim.x + threadIdx.x;
  if (e >= E) return;
  int s = src[e], d = dst[e], p = prop[e];
  int base = (lab[s] * LBL + lab[d]) * PRP + p;
  float4 w;
  w.x = W[base];
  w.y = W[base + LLP];
  w.z = W[base + 2 * LLP];
  w.w = W[base + 3 * LLP];
  uint32_t pos = atomicAdd(&cursor[s], 1u);
  sdst[pos] = d;
  sw4[pos] = w;
}

// ---------- gather/accumulate: one wave32 per node, WMMA f32 16x16x4 ----------
__global__ __launch_bounds__(256) void k_gather(const float* __restrict__ x,
                                                const int* __restrict__ lab,
                                                const float* __restrict__ Bias,
                                                const uint32_t* __restrict__ offs,
                                                const int* __restrict__ sdst,
                                                const float4* __restrict__ sw4,
                                                float* __restrict__ out, int N) {
  __shared__ float4 ls_w[8][2][32];
  __shared__ int    ls_d[8][2][32];

  const int lane = threadIdx.x & 31;
  const int wave = threadIdx.x >> 5;
  const int half = lane >> 4;       // 0: lanes 0-15, 1: lanes 16-31
  const int lsub = lane & 15;
  const int v = blockIdx.x * 8 + wave;
  if (v >= N) return;               // wave-uniform branch (EXEC stays all-1s)

  const uint32_t beg = offs[v];
  const uint32_t end = offs[v + 1];
  const int lb = lab[v];

  // 8 feature tiles of 16; C/D rows 0..3 (lanes 0-15) = out[k, fb..fb+15].
  // Seed rows 0..3 with bias, everything else with 0.
  v8f acc[8];
#pragma unroll
  for (int t = 0; t < 8; ++t) {
    v8f z = {0.f, 0.f, 0.f, 0.f, 0.f, 0.f, 0.f, 0.f};
#pragma unroll
    for (int k = 0; k < KCH; ++k)
      z[k] = (half == 0) ? Bias[(k * LBL + lb) * FDIM + t * 16 + lsub] : 0.f;
    acc[t] = z;
  }

  const uint32_t n = end - beg;
  const uint32_t nchunks = (n + 31u) >> 5;

  auto issue = [&](uint32_t c, int buf) {
    uint32_t j = beg + (c << 5) + (uint32_t)lane;
    if (j < end) {
      async_load_lds_b128(lds_addr_of(&ls_w[wave][buf][lane]), &sw4[j]);
      async_load_lds_b32(lds_addr_of(&ls_d[wave][buf][lane]), &sdst[j]);
    }
  };

  if (nchunks > 0) issue(0, 0);
  if (nchunks > 1) issue(1, 1);

  for (uint32_t c = 0; c < nchunks; ++c) {
    const int buf = (int)(c & 1u);
    if (c + 1 < nchunks) wait_async_le2(); else wait_async_le0();

    const uint32_t cbeg = beg + (c << 5);
    const uint32_t rem = end - cbeg;
    const uint32_t m = rem < 32u ? rem : 32u;
    const uint32_t quads = (m + 3u) >> 2;

    for (uint32_t q = 0; q < quads; ++q) {
      // kk mapping for both A and B: VGPR0 <-> kk = 2*half, VGPR1 <-> kk = 1+2*half
      const int e0 = (int)(q << 2) + 2 * half;      // edge slot for .x
      const int e1 = e0 + 1;                        // edge slot for .y
      const bool v0 = (cbeg + (uint32_t)e0) < end;
      const bool v1 = (cbeg + (uint32_t)e1) < end;
      const int e0c = v0 ? e0 : 0;                  // slot 0 is always valid in a chunk
      const int e1c = v1 ? e1 : 0;

      // A: lane lsub<4 carries channel lsub of w4[edge]; rows 4..15 zero.
      const float* w0 = (const float*)&ls_w[wave][buf][e0c];
      const float* w1 = (const float*)&ls_w[wave][buf][e1c];
      v2f a;
      a.x = (lsub < 4 && v0) ? w0[lsub] : 0.f;
      a.y = (lsub < 4 && v1) ? w1[lsub] : 0.f;

      // B: lane lsub carries feature fb+lsub of x[dst[edge]].
      const int d0 = ls_d[wave][buf][e0c];
      const int d1 = ls_d[wave][buf][e1c];
      const float* xr0 = x + (size_t)(uint32_t)d0 * FDIM + lsub;
      const float* xr1 = x + (size_t)(uint32_t)d1 * FDIM + lsub;

#pragma unroll
      for (int t = 0; t < 8; ++t) {
        v2f b;
        b.x = xr0[t * 16];
        b.y = xr1[t * 16];
        acc[t] = __builtin_amdgcn_wmma_f32_16x16x4_f32(
            /*neg_a=*/false, a, /*neg_b=*/false, b,
            /*c_mod=*/(short)0, acc[t], /*reuse_a=*/false, /*reuse_b=*/false);
      }
    }
    if (c + 2 < nchunks) issue(c + 2, buf);
  }

  // Rows 0..3 on lanes 0-15 hold out[k, t*16 + lsub].
  if (half == 0) {
#pragma unroll
    for (int t = 0; t < 8; ++t)
#pragma unroll
      for (int k = 0; k < KCH; ++k)
        out[((size_t)v * KCH + k) * FDIM + t * 16 + lsub] = acc[t][k];
  }
}

// ---------- fallback path (scratch too small): bias init + f32 atomic scatter ----------
__global__ void k_bias(const int* __restrict__ lab, const float4* __restrict__ B4,
                       float4* __restrict__ out4, int N) {
  int i = blockIdx.x * blockDim.x + threadIdx.x;
  if (i >= N * KCH * FV4) return;
  int f4 = i & 31, k = (i >> 5) & 3, v = i >> 7;
  out4[i] = B4[(k * LBL + lab[v]) * FV4 + f4];
}

__global__ void k_edge_atomic(const int* __restrict__ src, const int* __restrict__ dst,
                              const int* __restrict__ prop, const int* __restrict__ lab,
                              const float* __restrict__ W, const float4* __restrict__ x4,
                              float* __restrict__ out, int E) {
  int wid = blockIdx.x * (blockDim.x >> 5) + (threadIdx.x >> 5);
  int lane = threadIdx.x & 31;
  if (wid >= E) return;
  int s = src[wid], d = dst[wid], p = prop[wid];
  int base = (lab[s] * LBL + lab[d]) * PRP + p;
  float4 xv = x4[(uint32_t)d * FV4 + lane];
  for (int k = 0; k < KCH; ++k) {
    float w = W[base + k * LLP];
    float* o = &out[((size_t)s * KCH + k) * FDIM + lane * 4];
    atomicAdd(o + 0, w * xv.x);
    atomicAdd(o + 1, w * xv.y);
    atomicAdd(o + 2, w * xv.z);
    atomicAdd(o + 3, w * xv.w);
  }
}

extern "C" void kernel_launch(void* const* d_in, const int* in_sizes, int n_in,
                              void* d_out, int out_size, void* d_ws, size_t ws_size,
                              hipStream_t stream) {
  const float* x   = (const float*)d_in[0];
  const int* lab   = (const int*)d_in[1];
  const int* src   = (const int*)d_in[2];
  const int* dst   = (const int*)d_in[3];
  const int* prop  = (const int*)d_in[4];
  const float* W   = (const float*)d_in[5];
  const float* B   = (const float*)d_in[6];
  const int N = in_sizes[1];
  const int E = in_sizes[2];

  // Scratch layout: [ sw4: E float4 | sdst: E int | cnt: N | offs: N+1 | cursor: N ]
  const size_t need = (size_t)E * 20 + (size_t)(3 * N + 2) * 4;

  if (ws_size >= need) {
    char* ws = (char*)d_ws;
    float4*   sw4    = (float4*)ws;
    int*      sdst   = (int*)(ws + (size_t)E * 16);
    uint32_t* cnt    = (uint32_t*)(ws + (size_t)E * 20);
    uint32_t* offs   = cnt + N;
    uint32_t* cursor = offs + (N + 1);

    k_zero_u32<<<(N + 255) / 256, 256, 0, stream>>>(cnt, N);
    k_hist<<<(E + 255) / 256, 256, 0, stream>>>(src, cnt, E);
    k_scan<<<1, 1024, 0, stream>>>(cnt, offs, cursor, N, E);
    k_scatter<<<(E + 255) / 256, 256, 0, stream>>>(src, dst, prop, lab, W, cursor, sdst, sw4, E);
    k_gather<<<(N + 7) / 8, 256, 0, stream>>>(x, lab, B, offs, sdst, sw4, (float*)d_out, N);
  } else {
    k_bias<<<(N * KCH * FV4 + 255) / 256, 256, 0, stream>>>(lab, (const float4*)B,
                                                            (float4*)d_out, N);
    k_edge_atomic<<<(E + 7) / 8, 256, 0, stream>>>(src, dst, prop, lab, W,
                                                   (const float4*)x, (float*)d_out, E);
  }
}